// AdversarialLogLikelihoodLossLayer_48515950576448
// MI455X (gfx1250) — compile-verified
//
#include <hip/hip_runtime.h>

// Problem constants (from the reference)
#define BATCH 64
#define CH    40
#define NPROB 32000          // 64*500 independent 40x40 Cholesky+solve problems
#define RS    41             // row stride (floats) of y_pred tile AND of the LDS tile
#define WAVES 4              // waves (problems) per block
#define WSLOTS (48 * RS)     // 48 rows: rows 40..47 are WMMA scratch; 1968 floats

typedef __attribute__((ext_vector_type(2))) float v2f;
typedef __attribute__((ext_vector_type(8))) float v8f;
typedef int b128_t __attribute__((vector_size(16)));   // async-LDS builtin param type

#if defined(__has_builtin)
#if __has_builtin(__builtin_amdgcn_global_load_async_to_lds_b128)
#define USE_ASYNC_LDS 1
#endif
#endif

// Wave-local LDS fence: LDS ops are in-order per wave; we need compiler ordering
// (memory clobber) plus DScnt drained.
__device__ __forceinline__ void wave_sync() {
    asm volatile("s_wait_dscnt 0x0" ::: "memory");
}

__global__ void nll_init(float* out) { out[0] = 0.0f; }

// Register-resident panel factorization of cols [JB, JB+NB).
// Lane owns row 'lane' (p0) and row 32+(lane&7) (p1; lanes 8..31 hold identical
// duplicates). Column broadcasts use compile-time shuffle sources. Fully
// unguarded: above-diagonal entries turn into stale-but-finite values that are
// only ever consumed under a select mask downstream.
template <int JB, int NB>
__device__ __forceinline__ void factor_panel(float* __restrict__ L, int lane, int r1,
                                             float* __restrict__ invA, float& logdet) {
    float p0[NB], p1[NB];
    #pragma unroll
    for (int c = 0; c < NB; ++c) {
        p0[c] = L[lane * RS + JB + c];
        p1[c] = L[r1 * RS + JB + c];
    }
    #pragma unroll
    for (int j = 0; j < NB; ++j) {
        const int jc = JB + j;
        const float diag = __shfl((jc < 32) ? p0[j] : p1[j], jc & 31, 32);
        const float inv  = rsqrtf(diag);
        invA[jc] = inv;                       // compile-time index -> stays in VGPRs
        logdet += 0.5f * __logf(diag);        // log(L[j][j]) = 0.5*log(diag); replicated
        p0[j] *= inv;
        p1[j] *= inv;
        #pragma unroll
        for (int k = j + 1; k < NB; ++k) {
            const int kc = JB + k;
            const float bk = __shfl((kc < 32) ? p0[j] : p1[j], kc & 31, 32);
            p0[k] -= p0[j] * bk;
            p1[k] -= p1[j] * bk;
        }
    }
    #pragma unroll
    for (int c = 0; c < NB; ++c)
        L[lane * RS + JB + c] = p0[c];
    if (lane < 8) {
        #pragma unroll
        for (int c = 0; c < NB; ++c)
            L[r1 * RS + JB + c] = p1[c];
    }
}

// 16x16 trailing tile at (r0,c0): A -= P * P^T over panel cols jb..jb+15 via
// 4x V_WMMA_F32_16X16X4_F32 (B pre-negated; f32 WMMA has C-neg only).
// CLIP: restrict write-back to cols < c0+8 (the (32,32) corner: keeps every
// store in cols 32..39, so col 40 (mu) and stride-41 aliasing are never hit).
template <bool CLIP>
__device__ __forceinline__ void syrk_tile(float* __restrict__ L, int r0, int c0,
                                          int jb, int lane) {
    const int lhi = lane >> 4, llo = lane & 15;
    v8f acc;
    #pragma unroll
    for (int v = 0; v < 8; ++v)
        acc[v] = L[(r0 + v + 8 * lhi) * RS + c0 + llo];
    #pragma unroll
    for (int k = 0; k < 16; k += 4) {
        v2f a, b;
        const int ak = jb + k + 2 * lhi;          // lanes 0-15: K k,k+1 ; 16-31: k+2,k+3
        const int ar = (r0 + llo) * RS + ak;      // A fragment: M = llo
        a[0] = L[ar];
        a[1] = L[ar + 1];
        const int br = (c0 + llo) * RS + ak;      // B = -P^T : N = llo
        b[0] = -L[br];
        b[1] = -L[br + 1];
        acc = __builtin_amdgcn_wmma_f32_16x16x4_f32(false, a, false, b,
                                                    (short)0, acc, false, false);
    }
    if (!CLIP || llo < 8) {
        #pragma unroll
        for (int v = 0; v < 8; ++v)
            L[(r0 + v + 8 * lhi) * RS + c0 + llo] = acc[v];
    }
}

__global__ __launch_bounds__(WAVES * 32)
void nll_kernel(const float* __restrict__ y_true,
                const float* __restrict__ y_pred,
                float* __restrict__ out)
{
    __shared__ __align__(16) float Ls[WAVES * WSLOTS];

    const int lane = threadIdx.x & 31;
    const int w    = threadIdx.x >> 5;
    const int p    = blockIdx.x * WAVES + w;   // grid sized exactly: p < 32000
    const int r1   = 32 + (lane & 7);

    float* L = &Ls[w * WSLOTS];

    // ---- Stream the 40x41 tile (sigma | mu) into LDS; LDS layout == memory layout.
    // 410 x b128: 12 unconditional issues + 1 guarded.
#if USE_ASYNC_LDS
    {
        __attribute__((address_space(1))) b128_t* gsrc =
            (__attribute__((address_space(1))) b128_t*)(void*)
            const_cast<float*>(y_pred + (size_t)p * (CH * RS));
        __attribute__((address_space(3))) b128_t* ldst =
            (__attribute__((address_space(3))) b128_t*)(void*)L;
        #pragma unroll
        for (int u = 0; u < 12; ++u)
            __builtin_amdgcn_global_load_async_to_lds_b128(gsrc + lane + 32 * u,
                                                           ldst + lane + 32 * u, 0, 0);
        if (lane < 26)
            __builtin_amdgcn_global_load_async_to_lds_b128(gsrc + lane + 384,
                                                           ldst + lane + 384, 0, 0);
        asm volatile("s_wait_asynccnt 0x0" ::: "memory");
    }
#else
    {
        const float4* src = reinterpret_cast<const float4*>(y_pred + (size_t)p * (CH * RS));
        float4*       dst = reinterpret_cast<float4*>(L);
        #pragma unroll
        for (int u = 0; u < 12; ++u)
            dst[lane + 32 * u] = src[lane + 32 * u];
        if (lane < 26)
            dst[lane + 384] = src[lane + 384];
        wave_sync();
    }
#endif

    // ---- diff = y_true - mu (mu = LDS col 40; never written by any later phase).
    const float* yt = y_true + (size_t)p * CH;
    float d0 = yt[lane] - L[lane * RS + CH];                      // rows 0..31
    float d1 = 0.0f;
    if (lane < 8) d1 = yt[32 + lane] - L[r1 * RS + CH];           // rows 32..39

    float invA[CH];
    float logdet = 0.0f;

    // ---- Blocked Cholesky: panels in registers, trailing updates on the WMMA unit.
    factor_panel<0, 16>(L, lane, r1, invA, logdet);
    wave_sync();
    syrk_tile<false>(L, 16, 16, 0, lane);   // A[16:32,16:32] -= P0 P0^T
    syrk_tile<false>(L, 32, 16, 0, lane);   // A[32:48,16:32] -= P1 P0^T (rows 40+ scratch)
    syrk_tile<true >(L, 32, 32, 0, lane);   // A[32:48,32:40] -= P1 P1^T, k=0..15 (clipped)
    wave_sync();
    factor_panel<16, 16>(L, lane, r1, invA, logdet);
    wave_sync();
    syrk_tile<true >(L, 32, 32, 16, lane);  // corner, k=16..31 (clipped)
    wave_sync();
    factor_panel<32, 8>(L, lane, r1, invA, logdet);
    wave_sync();

    // ---- Forward substitution, fully unrolled; no divisions, select-masked FMAs.
    float sz2 = 0.0f;
    #pragma unroll
    for (int j = 0; j < CH; ++j) {
        const float dj = __shfl((j < 32) ? d0 : d1, j & 31, 32);
        const float zj = dj * invA[j];
        sz2 += zj * zj;
        if (j < 31) {                                  // some lane still has row > j
            const float lv0 = L[lane * RS + j];        // unconditional load; masked below
            d0 = (lane > j) ? d0 - lv0 * zj : d0;
        }
        const float lv1 = L[r1 * RS + j];
        d1 = (r1 > j) ? d1 - lv1 * zj : d1;            // j < 32: folds to plain FMA
    }

    if (lane == 0) {
        const float LOG2PI = 1.83787706640934548356f;
        const float ll = -0.5f * sz2 - logdet - 0.5f * (float)CH * LOG2PI;
        atomicAdd(out, -ll * (1.0f / (float)BATCH));   // out = -(1/B) * sum ll
    }
}

extern "C" void kernel_launch(void* const* d_in, const int* in_sizes, int n_in,
                              void* d_out, int out_size, void* d_ws, size_t ws_size,
                              hipStream_t stream)
{
    const float* y_true = (const float*)d_in[0];   // (64,500,40)
    const float* y_pred = (const float*)d_in[1];   // (64,500,40,41)
    float* out = (float*)d_out;                    // scalar f32

    nll_init<<<1, 1, 0, stream>>>(out);
    nll_kernel<<<NPROB / WAVES, WAVES * 32, 0, stream>>>(y_true, y_pred, out);
}